// MultiBoxLoss_4990751998352
// MI455X (gfx1250) — compile-verified
//
#include <hip/hip_runtime.h>
#include <hip/hip_bf16.h>

typedef float v2f __attribute__((ext_vector_type(2)));
typedef float v8f __attribute__((ext_vector_type(8)));

#define ONUM 16
#define CONF_BLOCKS 2048
#define EPSF 1e-6f

// ---------------------------------------------------------------------------
// Kernel 1: per-batch matching + positive loc loss. One block per batch image.
// ---------------------------------------------------------------------------
__global__ void match_kernel(const float* __restrict__ boxes,      // [B,O,4] xy
                             const int*   __restrict__ labels,     // [B,O]
                             const float* __restrict__ dboxes,     // [N,4] cxcywh
                             const float* __restrict__ locs_pred,  // [B,N,4]
                             float* __restrict__ best_iou,         // [B,N]
                             int*   __restrict__ best_obj,         // [B,N]
                             int*   __restrict__ t_class,          // [B,N]
                             float* __restrict__ loc_partial,      // [B]
                             int*   __restrict__ n_pos_arr,        // [B]
                             int N) {
  const int b   = blockIdx.x;
  const int tid = threadIdx.x;
  const int nth = blockDim.x;           // 256
  const int wave = tid >> 5;

  __shared__ float s_bx[ONUM][4];
  __shared__ float s_area[ONUM];
  __shared__ int   s_lab[ONUM];
  __shared__ float s_wv[8][ONUM];
  __shared__ int   s_wi[8][ONUM];
  __shared__ int   s_dbo[ONUM];

  if (tid < ONUM * 4) s_bx[tid >> 2][tid & 3] = boxes[(size_t)b * ONUM * 4 + tid];
  if (tid < ONUM)     s_lab[tid] = labels[b * ONUM + tid];
  __syncthreads();
  if (tid < ONUM)
    s_area[tid] = (s_bx[tid][2] - s_bx[tid][0]) * (s_bx[tid][3] - s_bx[tid][1]);
  __syncthreads();

  float vmax[ONUM]; int vidx[ONUM];
#pragma unroll
  for (int o = 0; o < ONUM; ++o) { vmax[o] = -1.0f; vidx[o] = 0x7FFFFFFF; }

  // Pass 1: per-anchor best object; per-object running best anchor.
  for (int n = tid; n < N; n += nth) {
    float dcx = dboxes[n * 4 + 0], dcy = dboxes[n * 4 + 1];
    float dw  = dboxes[n * 4 + 2], dh  = dboxes[n * 4 + 3];
    float dx0 = dcx - dw * 0.5f, dy0 = dcy - dh * 0.5f;
    float dx1 = dcx + dw * 0.5f, dy1 = dcy + dh * 0.5f;
    float darea = (dx1 - dx0) * (dy1 - dy0);
    float bestv = -1.0f; int besto = 0;
#pragma unroll
    for (int o = 0; o < ONUM; ++o) {
      float ltx = fmaxf(s_bx[o][0], dx0), lty = fmaxf(s_bx[o][1], dy0);
      float rbx = fminf(s_bx[o][2], dx1), rby = fminf(s_bx[o][3], dy1);
      float w = fmaxf(rbx - ltx, 0.0f),  h = fmaxf(rby - lty, 0.0f);
      float inter = w * h;
      float iou = inter / (s_area[o] + darea - inter);
      if (iou > bestv) { bestv = iou; besto = o; }                 // first-max
      if (iou > vmax[o] || (iou == vmax[o] && n < vidx[o])) { vmax[o] = iou; vidx[o] = n; }
    }
    best_iou[(size_t)b * N + n] = bestv;
    best_obj[(size_t)b * N + n] = besto;
  }

  // Reduce per-object best anchor across wave, then across waves.
#pragma unroll
  for (int o = 0; o < ONUM; ++o) {
    float v = vmax[o]; int ix = vidx[o];
    for (int off = 16; off; off >>= 1) {
      float ov = __shfl_xor(v, off);
      int   oi = __shfl_xor(ix, off);
      if (ov > v || (ov == v && oi < ix)) { v = ov; ix = oi; }
    }
    if ((tid & 31) == 0) { s_wv[wave][o] = v; s_wi[wave][o] = ix; }
  }
  __syncthreads();
  if (tid < ONUM) {
    float v = s_wv[0][tid]; int ix = s_wi[0][tid];
    for (int w = 1; w < 8; ++w) {
      float ov = s_wv[w][tid]; int oi = s_wi[w][tid];
      if (ov > v || (ov == v && oi < ix)) { v = ov; ix = oi; }
    }
    s_dbo[tid] = ix;
  }
  __syncthreads();
  // Override (sequential: scatter "last index wins" semantics).
  if (tid == 0) {
    for (int o = 0; o < ONUM; ++o) {
      int ns = s_dbo[o];
      best_obj[(size_t)b * N + ns] = o;
      best_iou[(size_t)b * N + ns] = 1.0f;
    }
  }
  __threadfence_block();
  __syncthreads();

  // Pass 2: labels, positive SmoothL1 loc loss.
  float loc_sum = 0.0f; int pos_cnt = 0;
  for (int n = tid; n < N; n += nth) {
    float iou = best_iou[(size_t)b * N + n];
    int   obj = best_obj[(size_t)b * N + n];
    int   lab = (iou < 0.5f) ? 0 : s_lab[obj];
    t_class[(size_t)b * N + n] = lab;
    if (lab != 0) {
      pos_cnt++;
      float x0 = s_bx[obj][0], y0 = s_bx[obj][1];
      float x1 = s_bx[obj][2], y1 = s_bx[obj][3];
      float bcx = 0.5f * (x0 + x1), bcy = 0.5f * (y0 + y1);
      float bw = x1 - x0, bh = y1 - y0;
      float dcx = dboxes[n * 4 + 0], dcy = dboxes[n * 4 + 1];
      float dw  = dboxes[n * 4 + 2], dh  = dboxes[n * 4 + 3];
      float g[4];
      g[0] = (bcx - dcx) / (dw * 0.1f + EPSF);
      g[1] = (bcy - dcy) / (dh * 0.1f + EPSF);
      g[2] = logf(bw / dw + EPSF) * 5.0f;
      g[3] = logf(bh / dh + EPSF) * 5.0f;
      const float* lp = locs_pred + ((size_t)b * N + n) * 4;
#pragma unroll
      for (int k = 0; k < 4; ++k) {
        float d = fabsf(lp[k] - g[k]);
        loc_sum += (d < 1.0f) ? 0.5f * d * d : d - 0.5f;
      }
    }
  }
  for (int off = 16; off; off >>= 1) {
    loc_sum += __shfl_xor(loc_sum, off);
    pos_cnt += __shfl_xor(pos_cnt, off);
  }
  if ((tid & 31) == 0) { s_wv[wave][0] = loc_sum; s_wi[wave][0] = pos_cnt; }
  __syncthreads();
  if (tid == 0) {
    float ls = 0.0f; int pc = 0;
    for (int w = 0; w < 8; ++w) { ls += s_wv[w][0]; pc += s_wi[w][0]; }
    loc_partial[b] = ls;
    n_pos_arr[b]   = pc;
  }
}

// ---------------------------------------------------------------------------
// Kernel 2: streaming per-anchor cross entropy (wave-per-anchor).
// ---------------------------------------------------------------------------
__global__ void conf_kernel(const float* __restrict__ cls_pred,  // [B*N, C]
                            const int*   __restrict__ t_class,   // [B*N]
                            float* __restrict__ neg_conf,        // [B*N]
                            float* __restrict__ conf_partial,    // [CONF_BLOCKS]
                            int total, int C) {
  const int tid  = threadIdx.x;
  const int lane = tid & 31;
  const int wave = tid >> 5;
  const int wpb  = blockDim.x >> 5;
  const int gw   = blockIdx.x * wpb + wave;
  const int nwaves = gridDim.x * wpb;
  float acc = 0.0f;

  for (int i = gw; i < total; i += nwaves) {
    const float* x = cls_pred + (size_t)i * C;
    if (i + nwaves < total)
      __builtin_prefetch(cls_pred + (size_t)(i + nwaves) * C, 0, 1);
    float x0 = x[lane];
    float x1 = (lane + 32 < C) ? x[lane + 32] : -1e30f;
    float x2 = (lane + 64 < C) ? x[lane + 64] : -1e30f;
    float m = fmaxf(fmaxf(x0, x1), x2);
    for (int off = 16; off; off >>= 1) m = fmaxf(m, __shfl_xor(m, off));
    float e = expf(x0 - m) + expf(x1 - m) + expf(x2 - m);
    for (int off = 16; off; off >>= 1) e += __shfl_xor(e, off);
    float lse = logf(e) + m;
    int cls = t_class[i];                       // wave-uniform
    float xc;
    if (cls < 32)       xc = __shfl(x0, cls);
    else if (cls < 64)  xc = __shfl(x1, cls - 32);
    else                xc = __shfl(x2, cls - 64);
    float conf = lse - xc;
    float negv = (cls != 0) ? 0.0f : conf;
    if (cls != 0) acc += conf;
    if (lane == 0) neg_conf[i] = negv;
  }

  __shared__ float s_acc[8];
  if (lane == 0) s_acc[wave] = acc;
  __syncthreads();
  if (tid == 0) {
    float s = 0.0f;
    for (int w = 0; w < wpb; ++w) s += s_acc[w];
    conf_partial[blockIdx.x] = s;               // always written: no init needed
  }
}

// ---------------------------------------------------------------------------
// Kernel 3: exact top-K sum per batch via bisection on IEEE bit patterns
// (values are >= 0, so uint ordering == float ordering).
// ---------------------------------------------------------------------------
__global__ void topk_kernel(const float* __restrict__ neg_conf,   // [B,N]
                            const int*   __restrict__ n_pos_arr,  // [B]
                            float* __restrict__ hard_partial,     // [B]
                            int N) {
  const int b = blockIdx.x;
  const int tid = threadIdx.x;
  const float* v = neg_conf + (size_t)b * N;
  int K = n_pos_arr[b] * 3;
  if (K > N) K = N;

  __shared__ int   s_cnt;
  __shared__ float s_sum;

  if (K <= 0) { if (tid == 0) hard_partial[b] = 0.0f; return; }

  long long lo = -1, hi = 0x7F800000LL;   // cnt(lo)=N>=K, cnt(hi)=0<K
  while (hi - lo > 1) {
    long long mid = (lo + hi) >> 1;
    unsigned t = (unsigned)mid;
    if (tid == 0) s_cnt = 0;
    __syncthreads();
    int c = 0;
    for (int n = tid; n < N; n += blockDim.x)
      if (__float_as_uint(v[n]) > t) c++;
    atomicAdd(&s_cnt, c);
    __syncthreads();
    int cnt = s_cnt;
    if (cnt >= K) lo = mid; else hi = mid;
    __syncthreads();
  }
  // hi = K-th largest value's bit pattern.
  float thr = __uint_as_float((unsigned)hi);
  if (tid == 0) { s_cnt = 0; s_sum = 0.0f; }
  __syncthreads();
  int c = 0; float s = 0.0f;
  for (int n = tid; n < N; n += blockDim.x) {
    float val = v[n];
    if (__float_as_uint(val) > (unsigned)hi) { c++; s += val; }
  }
  atomicAdd(&s_cnt, c);
  atomicAdd(&s_sum, s);
  __syncthreads();
  if (tid == 0) hard_partial[b] = s_sum + (float)(K - s_cnt) * thr;
}

// ---------------------------------------------------------------------------
// Kernel 4: finalize. Single wave; WMMA-ones row-sum reduction of the 2048
// conf partials (D[m][n] = sum_k A[m][k], accumulated over 32 issues).
// ---------------------------------------------------------------------------
__global__ void finalize_kernel(const float* __restrict__ conf_partial, int nConf,
                                const float* __restrict__ loc_partial,
                                const int*   __restrict__ n_pos_arr,
                                const float* __restrict__ hard_partial,
                                int B, float* __restrict__ out) {
  const int lane = threadIdx.x;   // blockDim.x == 32, EXEC all ones
  v8f c = {0.f, 0.f, 0.f, 0.f, 0.f, 0.f, 0.f, 0.f};
  v2f bones = {1.0f, 1.0f};
  for (int it = 0; it < nConf / 64; ++it) {
    int base = it * 64 + lane * 2;
    v2f a;
    a.x = conf_partial[base];
    a.y = conf_partial[base + 1];
    c = __builtin_amdgcn_wmma_f32_16x16x4_f32(
        /*neg_a=*/false, a, /*neg_b=*/false, bones,
        /*c_mod=*/(short)0, c, /*reuse_a=*/false, /*reuse_b=*/false);
  }
  // Column 0: rows 0-7 live in lane 0's 8 VGPRs, rows 8-15 in lane 16's.
  float srow = c[0] + c[1] + c[2] + c[3] + c[4] + c[5] + c[6] + c[7];
  float conf_total = __shfl(srow, 0) + __shfl(srow, 16);

  if (lane == 0) {
    float loc = 0.0f, hard = 0.0f; int np = 0;
    for (int b = 0; b < B; ++b) {
      loc  += loc_partial[b];
      hard += hard_partial[b];
      np   += n_pos_arr[b];
    }
    float npf = (float)np;
    float loc_loss  = loc / (npf * 4.0f);
    float conf_loss = (hard + conf_total) / npf;
    out[0] = 0.5f * loc_loss + conf_loss;     // ALPHA = 0.5
  }
}

// ---------------------------------------------------------------------------
extern "C" void kernel_launch(void* const* d_in, const int* in_sizes, int n_in,
                              void* d_out, int out_size, void* d_ws, size_t ws_size,
                              hipStream_t stream) {
  const float* locs_pred = (const float*)d_in[0];  // [B,N,4]
  const float* cls_pred  = (const float*)d_in[1];  // [B,N,C]
  const float* boxes     = (const float*)d_in[2];  // [B,O,4]
  const int*   labels    = (const int*)  d_in[3];  // [B,O]
  const float* dboxes    = (const float*)d_in[4];  // [N,4]

  const int N = in_sizes[4] / 4;
  const int B = in_sizes[3] / ONUM;
  const int C = (int)((long long)in_sizes[1] / ((long long)B * N));
  const size_t BN = (size_t)B * N;

  char* p = (char*)d_ws;
  float* best_iou  = (float*)p; p += BN * 4;
  int*   best_obj  = (int*)  p; p += BN * 4;
  int*   t_cls     = (int*)  p; p += BN * 4;
  float* negc      = (float*)p; p += BN * 4;
  float* loc_part  = (float*)p; p += (size_t)B * 4;
  int*   np_arr    = (int*)  p; p += (size_t)B * 4;
  float* hard_part = (float*)p; p += (size_t)B * 4;
  float* conf_part = (float*)p; p += (size_t)CONF_BLOCKS * 4;

  match_kernel<<<B, 256, 0, stream>>>(boxes, labels, dboxes, locs_pred,
                                      best_iou, best_obj, t_cls,
                                      loc_part, np_arr, N);
  conf_kernel<<<CONF_BLOCKS, 256, 0, stream>>>(cls_pred, t_cls, negc, conf_part,
                                               (int)BN, C);
  topk_kernel<<<B, 256, 0, stream>>>(negc, np_arr, hard_part, N);
  finalize_kernel<<<1, 32, 0, stream>>>(conf_part, CONF_BLOCKS,
                                        loc_part, np_arr, hard_part,
                                        B, (float*)d_out);
}